// Upsample_73701638799876
// MI455X (gfx1250) — compile-verified
//
#include <hip/hip_runtime.h>

namespace {
constexpr int H = 256, W = 256;     // input spatial dims
constexpr int OH = 512, OW = 512;   // output spatial dims (up=2)
constexpr int TIN = 32;             // input rows/cols covered per tile (-> 64x64 outputs)
constexpr int SROWS = 34, SCOLS = 34, SSTR = 36; // 34x34 tile + halo, padded stride
constexpr int NELEM = SROWS * SCOLS;             // 1156 staged elements

typedef float f4 __attribute__((ext_vector_type(4)));
}

__global__ __launch_bounds__(256) void upfirdn2d_up2_k4(
    const float* __restrict__ x, const float* __restrict__ kern,
    float* __restrict__ out)
{
    __shared__ float smem[SROWS * SSTR];

    const int tid = threadIdx.x;
    const int n  = blockIdx.z;          // image index (B*C)
    const int i0 = blockIdx.y * TIN;    // input tile origin
    const int j0 = blockIdx.x * TIN;

    // LDS byte offset of smem base (low 32 bits of the generic pointer)
    const unsigned lds0 = (unsigned)(unsigned long long)&smem[0];
    const unsigned long long xbase = (unsigned long long)x;
    const int imgoff = n * (H * W);

    // ---- Stage 34x34 input tile (zero halo) into LDS via CDNA5 async copy ----
    #pragma unroll
    for (int it = 0; it < 5; ++it) {
        const int e = tid + it * 256;
        if (e < NELEM) {
            const int r  = e / SCOLS;
            const int c  = e - r * SCOLS;
            const int gy = i0 - 1 + r;
            const int gx = j0 - 1 + c;
            const unsigned ldsb = lds0 + 4u * (unsigned)(r * SSTR + c);
            if ((unsigned)gy < (unsigned)H && (unsigned)gx < (unsigned)W) {
                const int voff = 4 * (imgoff + gy * W + gx);
                // GVS mode: saddr = uniform base, per-lane 32-bit offset, per-lane LDS addr
                asm volatile("global_load_async_to_lds_b32 %0, %1, %2"
                             :: "v"(ldsb), "v"(voff), "s"(xbase) : "memory");
            } else {
                smem[r * SSTR + c] = 0.0f;   // zero padding region
            }
        }
    }
    asm volatile("s_wait_asynccnt 0" ::: "memory");
    __syncthreads();

    // ---- Kernel weights (uniform loads; kw[a*4+b] = k2d[a][b]) ----
    float kw[16];
    #pragma unroll
    for (int t = 0; t < 16; ++t) kw[t] = kern[t];

    // Each thread: 8-wide x 2-high output patch from a 3x6 LDS window
    const int tx = tid & 7;      // 8 column groups (4 input cols each)
    const int ty = tid >> 3;     // 32 input rows
    const int lj = tx * 4;
    const int li = ty;

    float r0[6], r1[6], r2[6];
    #pragma unroll
    for (int k = 0; k < 6; ++k) {
        r0[k] = smem[(li + 0) * SSTR + lj + k];   // input row i-1
        r1[k] = smem[(li + 1) * SSTR + lj + k];   // input row i
        r2[k] = smem[(li + 2) * SSTR + lj + k];   // input row i+1
    }

    // Parity-selected 2x2 taps of the 4x4 kernel (true convolution, pad0=2)
    float oE[8], oO[8];
    #pragma unroll
    for (int k = 0; k < 4; ++k) {
        oE[2*k]   = kw[15]*r0[k]   + kw[13]*r0[k+1] + kw[7]*r1[k]   + kw[5]*r1[k+1];
        oE[2*k+1] = kw[14]*r0[k+1] + kw[12]*r0[k+2] + kw[6]*r1[k+1] + kw[4]*r1[k+2];
        oO[2*k]   = kw[11]*r1[k]   + kw[ 9]*r1[k+1] + kw[3]*r2[k]   + kw[1]*r2[k+1];
        oO[2*k+1] = kw[10]*r1[k+1] + kw[ 8]*r1[k+2] + kw[2]*r2[k+1] + kw[0]*r2[k+2];
    }

    const int oy = 2 * (i0 + li);
    const int ox = 2 * (j0 + lj);          // multiple of 8 -> 32B aligned stores
    float* po = out + ((size_t)n * OH + oy) * OW + ox;

    f4 vE0 = {oE[0], oE[1], oE[2], oE[3]};
    f4 vE1 = {oE[4], oE[5], oE[6], oE[7]};
    f4 vO0 = {oO[0], oO[1], oO[2], oO[3]};
    f4 vO1 = {oO[4], oO[5], oO[6], oO[7]};
    // write-once output stream: non-temporal, keep L2 for input halo reuse
    __builtin_nontemporal_store(vE0, (f4*)po);
    __builtin_nontemporal_store(vE1, (f4*)(po + 4));
    __builtin_nontemporal_store(vO0, (f4*)(po + OW));
    __builtin_nontemporal_store(vO1, (f4*)(po + OW + 4));
}

extern "C" void kernel_launch(void* const* d_in, const int* in_sizes, int n_in,
                              void* d_out, int out_size, void* d_ws, size_t ws_size,
                              hipStream_t stream) {
    const float* x    = (const float*)d_in[0];
    const float* kern = (const float*)d_in[1];
    float* out = (float*)d_out;
    const int N = in_sizes[0] / (H * W);   // B*C = 512 images
    dim3 grid(OW / (2 * TIN), OH / (2 * TIN), N);  // 8 x 8 x 512
    upfirdn2d_up2_k4<<<grid, dim3(256), 0, stream>>>(x, kern, out);
}